// HungarianMatcher_70317204570533
// MI455X (gfx1250) — compile-verified
//
#include <hip/hip_runtime.h>
#include <stdint.h>

typedef float v4f __attribute__((ext_vector_type(4)));
typedef int   v4i __attribute__((ext_vector_type(4)));

#define NC   92      // classes
#define MTOT 2048    // targets total
#define NTOT 32000   // bs*nq = 32*1000
#define QT   16      // queries per block
#define MT   1024    // targets per block
#define TPB  256
#define EPSF 1e-6f

// Low 32 bits of the shared-aperture flat address == LDS byte offset.
__device__ __forceinline__ uint32_t lds_byte_off(const void* p) {
  return (uint32_t)(uintptr_t)p;
}

__global__ __launch_bounds__(TPB)
void matcher_cost_kernel(const float* __restrict__ logits,   // [NTOT, 92]
                         const float* __restrict__ qboxes,   // [NTOT, 4] cxcywh
                         const int*   __restrict__ tids,     // [MTOT]
                         const float* __restrict__ tboxes,   // [MTOT, 4] cxcywh
                         float* __restrict__ out) {          // [NTOT, MTOT]
  __shared__ float s_prob[QT][96];   // softmax probs, padded 92->96
  __shared__ float s_q[QT][12];      // cx,cy,w,h, x0,y0,x1,y1, area
  __shared__ float s_tb[MT * 4];     // target cxcywh tile
  __shared__ int   s_tid[MT];        // target class ids tile

  const int tid = threadIdx.x;
  const int n0 = blockIdx.x * QT;
  const int m0 = blockIdx.y * MT;

  // ---- async stage target tile (bbox 16KB + ids 4KB) into LDS ----
  // Issued first so the DMA overlaps the softmax below (tracked by ASYNCcnt).
  {
    uint32_t lb = lds_byte_off(&s_tb[0]);
#pragma unroll
    for (int it = 0; it < 4; ++it) {
      int g = it * TPB + tid;                                    // 0..1023
      uint64_t ga = (uint64_t)(uintptr_t)(tboxes + (size_t)(m0 + g) * 4);
      uint32_t la = lb + (uint32_t)g * 16u;
      asm volatile("global_load_async_to_lds_b128 %0, %1, off"
                   :: "v"(la), "v"(ga) : "memory");
    }
    uint64_t gi = (uint64_t)(uintptr_t)(tids + m0 + tid * 4);
    uint32_t li = lds_byte_off(&s_tid[0]) + (uint32_t)tid * 16u;
    asm volatile("global_load_async_to_lds_b128 %0, %1, off"
                 :: "v"(li), "v"(gi) : "memory");
  }

  // ---- softmax: wave w handles query rows w and w+8 (wave32) ----
  const int wave = tid >> 5, lane = tid & 31;
  const float L2E = 1.4426950408889634f;
#pragma unroll
  for (int rr = 0; rr < 2; ++rr) {
    const int r = wave + rr * 8;
    const float* Lr = logits + (size_t)(n0 + r) * NC;
    float x0 = Lr[lane];
    float x1 = Lr[lane + 32];
    float x2 = (lane < NC - 64) ? Lr[lane + 64] : -1e30f;
    float mx = fmaxf(x0, fmaxf(x1, x2));
#pragma unroll
    for (int off = 16; off; off >>= 1) mx = fmaxf(mx, __shfl_xor(mx, off, 32));
    float e0 = __builtin_amdgcn_exp2f((x0 - mx) * L2E);
    float e1 = __builtin_amdgcn_exp2f((x1 - mx) * L2E);
    float e2 = (lane < NC - 64) ? __builtin_amdgcn_exp2f((x2 - mx) * L2E) : 0.0f;
    float s = e0 + e1 + e2;
#pragma unroll
    for (int off = 16; off; off >>= 1) s += __shfl_xor(s, off, 32);
    float inv = __builtin_amdgcn_rcpf(s);
    s_prob[r][lane]      = e0 * inv;
    s_prob[r][lane + 32] = e1 * inv;
    if (lane < NC - 64) s_prob[r][lane + 64] = e2 * inv;
  }

  // ---- per-query box precompute (xyxy + area) ----
  if (tid < QT) {
    v4f b = *(const v4f*)(qboxes + (size_t)(n0 + tid) * 4);
    float x0 = b.x - 0.5f * b.z, y0 = b.y - 0.5f * b.w;
    float x1 = b.x + 0.5f * b.z, y1 = b.y + 0.5f * b.w;
    s_q[tid][0] = b.x; s_q[tid][1] = b.y; s_q[tid][2] = b.z; s_q[tid][3] = b.w;
    s_q[tid][4] = x0;  s_q[tid][5] = y0;  s_q[tid][6] = x1;  s_q[tid][7] = y1;
    s_q[tid][8] = (x1 - x0) * (y1 - y0);
  }

  asm volatile("s_wait_asynccnt 0" ::: "memory");
  __syncthreads();

  // ---- each thread owns 4 consecutive targets: precompute into registers ----
  float tcx[4], tcy[4], tw[4], th_[4];
  float tx0[4], ty0[4], tx1[4], ty1[4], ta[4];
  v4i ids = *(const v4i*)&s_tid[tid * 4];
#pragma unroll
  for (int j = 0; j < 4; ++j) {
    v4f b = *(const v4f*)&s_tb[(tid * 4 + j) * 4];
    tcx[j] = b.x; tcy[j] = b.y; tw[j] = b.z; th_[j] = b.w;
    tx0[j] = b.x - 0.5f * b.z; ty0[j] = b.y - 0.5f * b.w;
    tx1[j] = b.x + 0.5f * b.z; ty1[j] = b.y + 0.5f * b.w;
    ta[j] = (tx1[j] - tx0[j]) * (ty1[j] - ty0[j]);
  }

  const int mbase = m0 + tid * 4;
#pragma unroll
  for (int q = 0; q < QT; ++q) {
    float qcx = s_q[q][0], qcy = s_q[q][1], qw = s_q[q][2], qh = s_q[q][3];
    float qx0 = s_q[q][4], qy0 = s_q[q][5], qx1 = s_q[q][6], qy1 = s_q[q][7];
    float qa  = s_q[q][8];
    v4f res;
#pragma unroll
    for (int j = 0; j < 4; ++j) {
      float p = s_prob[q][ids[j]];                         // class cost gather
      float l1 = fabsf(qcx - tcx[j]) + fabsf(qcy - tcy[j]) +
                 fabsf(qw  - tw[j])  + fabsf(qh  - th_[j]);
      float iw = fmaxf(fminf(qx1, tx1[j]) - fmaxf(qx0, tx0[j]), 0.0f);
      float ih = fmaxf(fminf(qy1, ty1[j]) - fmaxf(qy0, ty0[j]), 0.0f);
      float inter = iw * ih;
      float uni = qa + ta[j] - inter;
      float iou = inter * __builtin_amdgcn_rcpf(uni + EPSF);
      float ew = fmaxf(fmaxf(qx1, tx1[j]) - fminf(qx0, tx0[j]), 0.0f);
      float eh = fmaxf(fmaxf(qy1, ty1[j]) - fminf(qy0, ty0[j]), 0.0f);
      float ea = ew * eh;
      float giou = iou - (ea - uni) * __builtin_amdgcn_rcpf(ea + EPSF);
      res[j] = 5.0f * l1 - p - 2.0f * giou;
    }
    // Output (262MB) exceeds L2 (192MB): non-temporal 128-bit store keeps L2
    // free for the fully-resident inputs.
    float* dst = out + (size_t)(n0 + q) * MTOT + mbase;
    __builtin_nontemporal_store(res, (v4f*)dst);
  }
}

extern "C" void kernel_launch(void* const* d_in, const int* in_sizes, int n_in,
                              void* d_out, int out_size, void* d_ws, size_t ws_size,
                              hipStream_t stream) {
  const float* logits = (const float*)d_in[0];   // [32,1000,92]
  const float* qboxes = (const float*)d_in[1];   // [32,1000,4]
  const int*   tids   = (const int*)d_in[2];     // [2048]
  const float* tboxes = (const float*)d_in[3];   // [2048,4]
  float* out = (float*)d_out;                    // [32,1000,2048]
  dim3 grid(NTOT / QT, MTOT / MT);               // (2000, 2)
  matcher_cost_kernel<<<grid, TPB, 0, stream>>>(logits, qboxes, tids, tboxes, out);
}